// CrankNicolsonSolver_22299470201648
// MI455X (gfx1250) — compile-verified
//
#include <hip/hip_runtime.h>
#include <hip/hip_bf16.h>

// Crank-Nicolson batched solver for MI455X (gfx1250).
// Kernel 1: 1 workgroup (512 thr, 16 wave32) per batch element (128 WGs).
//   - sigma staged by the Tensor Data Mover (tensor_load_to_lds) in 16-step
//     chunks, double buffered, TDM LDS-padding => conflict-free 17-dword rows.
//   - 256 time steps, each a 9-level parallel cyclic reduction in LDS.
//   - writes V0 rows (513 + 7 zero pad, stride 520) to d_ws.
// Kernel 2: out(128x128) = V0(128x513) . W(513x128), W = interpolation
//   operator (2 nonzeros/column), computed with v_wmma_f32_16x16x4_f32,
//   one 16x16 tile per wave, K accumulated in steps of 4.

#define RR   0.03f
#define QQ   0.01f
#define MM   512     // spatial intervals (M)
#define NN   256     // time steps (N)
#define NTH  512     // threads per workgroup (= PCR system size incl. 1 pad row)
#define CH   16      // time steps staged per chunk
#define SBS  17      // LDS row stride in dwords (16 data + 1 TDM pad, odd => no conflicts)
#define WSS  520     // V0 scratch row stride (513 data + 7 zero pad, mult of 4)

typedef unsigned int u32x4 __attribute__((ext_vector_type(4)));
typedef int          i32x4 __attribute__((ext_vector_type(4)));
typedef int          i32x8 __attribute__((ext_vector_type(8)));
typedef float        v2f   __attribute__((ext_vector_type(2)));
typedef float        v8f   __attribute__((ext_vector_type(8)));

#if __has_builtin(__builtin_amdgcn_tensor_load_to_lds) && \
    __has_builtin(__builtin_amdgcn_s_wait_tensorcnt)
#define USE_TDM 1
#else
#define USE_TDM 0
#endif

#if __has_builtin(__builtin_amdgcn_s_wait_asynccnt)
#define WAIT_ASYNC(n) __builtin_amdgcn_s_wait_asynccnt(n)
#else
#define WAIT_ASYNC(n) asm volatile("s_wait_asynccnt %0" :: "i"(n) : "memory")
#endif

__device__ __forceinline__ unsigned lds_off(const void* p) {
  return (unsigned)(size_t)(__attribute__((address_space(3))) const void*)p;
}

#if USE_TDM
// One TDM descriptor moves the whole 511x16 f32 tile (row stride 257 floats)
// from global sigma into LDS, adding 1 dword of padding per 16-dword row.
__device__ __forceinline__ void tdm_issue(const float* gbase, unsigned dst_lds) {
  const unsigned long long ga = (unsigned long long)(size_t)gbase;
  const u32x4 g0 = {
      1u,                                                  // count=1 (valid D#)
      dst_lds,                                             // lds_addr (bytes)
      (unsigned)(ga & 0xFFFFFFFFu),                        // global_addr[31:0]
      (unsigned)((ga >> 32) & 0x01FFFFFFull) | (2u << 30)  // addr[56:32] | type=2
  };
  const i32x8 g1 = {
      (int)((2u << 16) | (1u << 20) | (3u << 22)),  // data_size=4B, pad_en, intv=16dw (amount=1dw)
      (int)(16u << 16),                             // tensor_dim0 = 16 (bits 79:48)
      (int)(511u << 16),                            // tensor_dim1 = 511 (bits 111:80)
      (int)(16u << 16),                             // tile_dim0 = 16 (bits 127:112)
      511,                                          // tile_dim1 = 511 (bits 143:128)
      257,                                          // tensor_dim0_stride lo32 (bits 207:160)
      0, 0                                          // stride hi, dim1_stride (unused, 2D)
  };
  const i32x4 z4 = {0, 0, 0, 0};
#if __clang_major__ >= 23
  const i32x8 z8 = {0, 0, 0, 0, 0, 0, 0, 0};
  __builtin_amdgcn_tensor_load_to_lds(g0, g1, z4, z4, z8, 0);
#else
  __builtin_amdgcn_tensor_load_to_lds(g0, g1, z4, z4, 0);
#endif
}
#else
__device__ __forceinline__ void async_ld_b32(const float* g, float* l) {
#if __has_builtin(__builtin_amdgcn_global_load_async_to_lds_b32)
  __builtin_amdgcn_global_load_async_to_lds_b32(
      (__attribute__((address_space(1))) int*)g,
      (__attribute__((address_space(3))) int*)l, 0, 0);
#else
  unsigned lo_ = lds_off(l);
  asm volatile("global_load_async_to_lds_b32 %0, %1, off"
               :: "v"(lo_), "v"(g) : "memory");
#endif
}
#endif

__global__ __launch_bounds__(NTH) void cn_pcr_kernel(
    const float* __restrict__ S0, const float* __restrict__ K,
    const float* __restrict__ T,  const float* __restrict__ sigma,
    float* __restrict__ out, float* __restrict__ ws, int write_out)
{
  __shared__ float4 eq[2][NTH];            // 16 KB : PCR (L,D,U,R) double buffer
  __shared__ float  V[MM + 8];             //  2 KB : V[0..512]
  __shared__ float  sbuf[2][NTH * SBS];    // 68 KB : sigma staging, padded rows

  const int b   = blockIdx.x;
  const int tid = threadIdx.x;

  const float S0b  = S0[b];
  const float Kb   = K[b];
  const float Tv   = T[0];
  const float Smax = 3.0f * fmaxf(S0b, Kb);
  const float dt   = Tv / (float)NN;

  // Terminal payoff: V_T[i] = max(Smax*i/M - K, 0)
  for (int i = tid; i <= MM; i += NTH)
    V[i] = fmaxf(Smax * ((float)i * (1.0f / (float)MM)) - Kb, 0.0f);

  // Row this thread owns: rows 0..510 real, row 511 identity padding.
  // Row j <-> grid index i=j+1; sigma row = sigma[b, j+1, :].
  const int   jj     = (tid < MM - 1) ? tid : (MM - 2);
  const float Si     = Smax * ((float)(jj + 1) * (1.0f / (float)MM));
  const float dS     = Smax * (1.0f / (float)MM);
  const float ratio2 = (Si * Si) / (dS * dS);
  const float drift  = ((RR - QQ) * Si) / dS;

  const float* sig_base =
      sigma + ((size_t)b * (MM + 1) + 1) * (NN + 1);     // sigma[b, 1, 0]
#if !USE_TDM
  const float* sigrow = sig_base + (size_t)jj * (NN + 1);
#endif
  const bool issuer = (tid < 32);                        // wave 0 drives the TDM

  // Prefetch first chunk (time steps 240..255) into sbuf[0].
#if USE_TDM
  if (issuer) tdm_issue(sig_base + (NN - CH), lds_off(&sbuf[0][0]));
#else
  for (int k = 0; k < CH; ++k)
    async_ld_b32(sigrow + (NN - CH) + k, &sbuf[0][tid * SBS + k]);
#endif

  int p = 0;
  for (int n0 = NN - CH; n0 >= 0; n0 -= CH) {
    // Issue next (earlier-in-time) chunk, then wait for the current one.
    // TDM/async ops complete in order per wave, so "<= 1 chunk outstanding"
    // guarantees the chunk we are about to read is in LDS.
#if USE_TDM
    if (n0 > 0 && issuer) tdm_issue(sig_base + (n0 - CH), lds_off(&sbuf[p ^ 1][0]));
    if (issuer) {
      if (n0 > 0) __builtin_amdgcn_s_wait_tensorcnt(1);
      else        __builtin_amdgcn_s_wait_tensorcnt(0);
    }
#else
    if (n0 > 0) {
      for (int k = 0; k < CH; ++k)
        async_ld_b32(sigrow + (n0 - CH) + k, &sbuf[p ^ 1][tid * SBS + k]);
      WAIT_ASYNC(CH);
    } else {
      WAIT_ASYNC(0);
    }
#endif
    asm volatile("" ::: "memory");
    __syncthreads();   // publish staged sigma (and initial V) to all waves

    for (int k = CH - 1; k >= 0; --k) {        // time step n = n0 + k, descending
      const float sg   = sbuf[p][jj * SBS + k];
      const float diff = (sg * sg) * ratio2;
      const float aj   =  0.25f * dt * (diff - drift);
      const float bj   = -0.5f  * dt * (diff + RR);
      const float cj   =  0.25f * dt * (diff + drift);

      // Row (L,D,U,R) of the CN system, exactly as the reference builds it.
      float L, D, U, R;
      if (tid < MM - 1) {
        float rhs = (1.0f + bj) * V[tid + 1] + aj * V[tid + 2] + cj * V[tid];
        if (tid == 0)      rhs += aj * V[0];
        if (tid == MM - 2) rhs += cj * V[MM];
        L = (tid == 0)      ? 0.0f : -aj;
        U = (tid == MM - 2) ? 0.0f : -cj;
        D = 1.0f - bj;
        R = rhs;
      } else {                                 // identity padding row
        L = 0.0f; D = 1.0f; U = 0.0f; R = 0.0f;
      }

      // ---- Parallel cyclic reduction: 9 levels over 512 rows ----
      int q = 0;
      eq[0][tid] = make_float4(L, D, U, R);
      __syncthreads();
#pragma unroll
      for (int s = 1; s < NTH; s <<= 1) {
        const float4 ident = make_float4(0.0f, 1.0f, 0.0f, 0.0f);
        const float4 em = (tid - s >= 0)  ? eq[q][tid - s] : ident;
        const float4 ep = (tid + s < NTH) ? eq[q][tid + s] : ident;
        const float alpha = L * __builtin_amdgcn_rcpf(em.y);
        const float gamma = U * __builtin_amdgcn_rcpf(ep.y);
        const float Ln = -alpha * em.x;
        const float Un = -gamma * ep.z;
        const float Dn = D - alpha * em.z - gamma * ep.x;
        const float Rn = R - alpha * em.w - gamma * ep.w;
        L = Ln; D = Dn; U = Un; R = Rn;
        eq[q ^ 1][tid] = make_float4(L, D, U, R);
        q ^= 1;
        __syncthreads();
      }
      const float x = R * __builtin_amdgcn_rcpf(D);

      // V = [0, x, 0]  (boundaries zeroed each step, as in reference)
      if (tid < MM - 1) V[tid + 1] = x;
      if (tid == MM - 1) { V[0] = 0.0f; V[MM] = 0.0f; }
      __syncthreads();
    }
    p ^= 1;
  }

  if (write_out) {
    // Fallback epilogue: direct gather interpolation.
    if (tid < 128) {
      const float pos = (float)tid * (float)(512.0 / 127.0);
      int lo = (int)floorf(pos);
      if (lo > MM) lo = MM;
      const int   hi = (lo + 1 < MM) ? (lo + 1) : MM;
      const float w  = pos - (float)lo;
      out[(size_t)b * 128 + tid] = V[lo] * (1.0f - w) + V[hi] * w;
    }
  } else {
    // Dump V0 row for the WMMA interpolation GEMM (zero-padded to 520).
    float* wsrow = ws + (size_t)b * WSS;
    wsrow[tid] = V[tid];
    if (tid == 0)              wsrow[MM] = V[MM];
    else if (tid < 8)          wsrow[MM + tid] = 0.0f;
  }
}

// out(128x128) = V0(128x513,pad520) . W(513x128) on the matrix engine.
// W[k,j] = (1-w_j)*[k==lo_j] + w_j*[k==hi_j], synthesized in registers.
// One wave per 16x16 output tile; K swept 4 at a time (130 steps over pad-520).
__global__ __launch_bounds__(32) void interp_wmma_kernel(
    const float* __restrict__ ws, float* __restrict__ out)
{
  const int lane = threadIdx.x;            // 0..31
  const int ti   = blockIdx.x >> 3;        // output row tile (batch dim)
  const int tj   = blockIdx.x & 7;         // output col tile (interp points)
  const int nloc = lane & 15;
  const int half = lane >> 4;              // 0: lanes 0-15, 1: lanes 16-31

  // Per-lane interpolation column parameters (column n = tj*16 + nloc).
  const int   n   = tj * 16 + nloc;
  const float pos = (float)n * (float)(512.0 / 127.0);
  int lo = (int)floorf(pos);
  if (lo > MM) lo = MM;
  const int   hi  = (lo + 1 < MM) ? (lo + 1) : MM;
  const float w   = pos - (float)lo;
  const float wlo = 1.0f - w;

#if __has_builtin(__builtin_amdgcn_wmma_f32_16x16x4_f32)
  // A 16x4 layout: lane<16 holds (M=lane, K=0/1) in v0/v1; lane>=16 K=2/3.
  // B 4x16 layout: lane<16 holds (K=0/1, N=lane) in v0/v1; lane>=16 K=2/3.
  // C/D: VGPR r -> M=r (lanes 0-15) / M=8+r (lanes 16-31), N=lane%16.
  const float* arow = ws + (size_t)(ti * 16 + nloc) * WSS;
  const int    koff = half << 1;           // 0 or 2
  v8f c = {0.f, 0.f, 0.f, 0.f, 0.f, 0.f, 0.f, 0.f};
  for (int kb = 0; kb < WSS / 4; ++kb) {
    const int k0 = kb * 4 + koff;
    v2f a; a.x = arow[k0]; a.y = arow[k0 + 1];
    v2f bb;
    bb.x = ((k0     == lo) ? wlo : 0.0f) + ((k0     == hi) ? w : 0.0f);
    bb.y = ((k0 + 1 == lo) ? wlo : 0.0f) + ((k0 + 1 == hi) ? w : 0.0f);
    c = __builtin_amdgcn_wmma_f32_16x16x4_f32(
        false, a, false, bb, (short)0, c, false, false);
  }
#pragma unroll
  for (int r = 0; r < 8; ++r) {
    const int m = r + (half << 3);
    out[(size_t)(ti * 16 + m) * 128 + n] = c[r];
  }
#else
  // Scalar fallback: identical result via direct gather.
#pragma unroll
  for (int r = 0; r < 8; ++r) {
    const int i = ti * 16 + r + (half << 3);
    const float* row = ws + (size_t)i * WSS;
    out[(size_t)i * 128 + n] = row[lo] * wlo + row[hi] * w;
  }
#endif
}

extern "C" void kernel_launch(void* const* d_in, const int* in_sizes, int n_in,
                              void* d_out, int out_size, void* d_ws, size_t ws_size,
                              hipStream_t stream) {
  // setup_inputs order: S0(128 f32), K(128 f32), T(1 f32),
  //                     sigma(128*513*257 f32), M(int), N(int)
  const float* S0    = (const float*)d_in[0];
  const float* K     = (const float*)d_in[1];
  const float* T     = (const float*)d_in[2];
  const float* sigma = (const float*)d_in[3];
  float* out = (float*)d_out;
  float* ws  = (float*)d_ws;

  const size_t ws_need = (size_t)128 * WSS * sizeof(float);
  const int direct = (ws != nullptr && ws_size >= ws_need) ? 0 : 1;

  cn_pcr_kernel<<<dim3(128), dim3(NTH), 0, stream>>>(S0, K, T, sigma,
                                                     out, ws, direct);
  if (!direct)
    interp_wmma_kernel<<<dim3(64), dim3(32), 0, stream>>>(ws, out);

  (void)in_sizes; (void)n_in; (void)out_size;
}